// TensorCircuit_64458869178386
// MI455X (gfx1250) — compile-verified
//
#include <hip/hip_runtime.h>
#include <math.h>

typedef float v2f __attribute__((ext_vector_type(2)));
typedef float v8f __attribute__((ext_vector_type(8)));

#define CHI   384
#define TWO_CHI 768
#define NQ    24
#define NLAY  4
#define SITE  (CHI*2*CHI)       // 294912 floats per site tensor
#define JNB   32                // persistent jacobi blocks (must be co-resident)
#define NSWEEP 8

// ---------------------------------------------------------------------------
// 1) Single-qubit gates:  out[n,a,i,b] = sum_j g1[n,i,j] * in[n,a,j,b]
// ---------------------------------------------------------------------------
__global__ void mps_gate1_kernel(const float* __restrict__ in, float* __restrict__ out,
                                 const float* __restrict__ g1, int total) {
    int idx = blockIdx.x * blockDim.x + threadIdx.x;
    if (idx >= total) return;                       // total = NQ*CHI*CHI
    int nq  = idx / (CHI * CHI);
    int rem = idx - nq * (CHI * CHI);
    int a   = rem / CHI;
    int b   = rem - a * CHI;
    const float* g = g1 + nq * 4;                   // [i][j] at i*2+j
    size_t base = (((size_t)nq * CHI + a) * 2) * CHI + b;
    float x0 = in[base];
    float x1 = in[base + CHI];
    out[base]       = g[0] * x0 + g[1] * x1;
    out[base + CHI] = g[2] * x0 + g[3] * x1;
}

// ---------------------------------------------------------------------------
// 2) Generic f32 WMMA GEMM: C(MxN) = A(MxK) * B(KxN), row-major, optional A^T
//    Block: 128 threads (4 waves), 64x64 tile; each wave 32x32 (2x2 WMMA tiles)
//    Compute: V_WMMA_F32_16X16X4_F32 (CDNA5 FP32 matrix op).
//    Staging: GLOBAL_LOAD_ASYNC_TO_LDS_B32/B128 (ASYNCcnt) -> s_wait_asynccnt.
//    Dynamic LDS only => dynamic segment starts at LDS offset 0, so raw LDS
//    byte offsets for the async VDST operand are plain arithmetic.
// ---------------------------------------------------------------------------
#define TM 64
#define TN 64
#define TK 32
#define AS_OFF(r, c) (((r) * (TK + 1) + (c)) * 4)
#define BS_OFF(r, c) ((TM * (TK + 1) + (r) * TN + (c)) * 4)
#define GEMM_LDS_BYTES ((TM * (TK + 1) + TK * TN) * 4)

__global__ __launch_bounds__(128)
void gemm_wmma_f32(const float* __restrict__ A, const float* __restrict__ B,
                   float* __restrict__ C, int M, int N, int K,
                   int lda, int ldb, int ldc, int transA) {
    extern __shared__ float smem[];          // [As: TM x (TK+1)][Bs: TK x TN]
    const int tid  = threadIdx.x;
    const int lane = tid & 31;
    const int wave = tid >> 5;                 // 0..3
    const int wr   = (wave >> 1) * 32;         // wave tile row in block tile
    const int wc   = (wave & 1) * 32;          // wave tile col
    const int br   = blockIdx.y * TM;
    const int bc   = blockIdx.x * TN;
    const int m16  = lane & 15;
    const int kb   = (lane >> 4) * 2;          // K pair base: lanes 0-15 -> K0/K1, 16-31 -> K2/K3
    const int rhalf = (lane >> 4) * 8;         // C/D: lanes>=16 hold rows M+8..M+15

    v8f acc[2][2];
#pragma unroll
    for (int i = 0; i < 2; i++)
#pragma unroll
        for (int j = 0; j < 2; j++) acc[i][j] = (v8f){0,0,0,0,0,0,0,0};

    for (int k0 = 0; k0 < K; k0 += TK) {
        // --- async stage A tile (scalar b32: supports transposed gather) ---
#pragma unroll
        for (int t = 0; t < (TM * TK) / 128; ++t) {
            int idx = t * 128 + tid;
            int rr = idx / TK, cc = idx % TK;
            const float* gp = transA ? (A + (size_t)(k0 + cc) * lda + (br + rr))
                                     : (A + (size_t)(br + rr) * lda + (k0 + cc));
            unsigned loff = (unsigned)AS_OFF(rr, cc);
            asm volatile("global_load_async_to_lds_b32 %0, %1, off"
                         :: "v"(loff), "v"(gp) : "memory");
        }
        // --- async stage B tile (b128: rows contiguous, 16B aligned) -------
#pragma unroll
        for (int t = 0; t < (TK * TN) / (128 * 4); ++t) {
            int idx = t * 128 + tid;           // 0..511 float4 slots
            int rr = idx / (TN / 4);
            int c4 = (idx % (TN / 4)) * 4;
            const float* gp = B + (size_t)(k0 + rr) * ldb + (bc + c4);
            unsigned loff = (unsigned)BS_OFF(rr, c4);
            asm volatile("global_load_async_to_lds_b128 %0, %1, off"
                         :: "v"(loff), "v"(gp) : "memory");
        }
        asm volatile("s_wait_asynccnt 0x0" ::: "memory");
        __syncthreads();

#pragma unroll
        for (int kk = 0; kk < TK; kk += 4) {
            v2f af[2], bf[2];
#pragma unroll
            for (int t = 0; t < 2; t++) {
                int r0 = wr + t * 16;
                af[t].x = smem[(r0 + m16) * (TK + 1) + kk + kb];
                af[t].y = smem[(r0 + m16) * (TK + 1) + kk + kb + 1];
                int c0 = wc + t * 16;
                bf[t].x = smem[TM * (TK + 1) + (kk + kb) * TN + c0 + m16];
                bf[t].y = smem[TM * (TK + 1) + (kk + kb + 1) * TN + c0 + m16];
            }
#pragma unroll
            for (int i = 0; i < 2; i++)
#pragma unroll
                for (int j = 0; j < 2; j++)
                    acc[i][j] = __builtin_amdgcn_wmma_f32_16x16x4_f32(
                        false, af[i], false, bf[j], (short)0, acc[i][j], false, false);
        }
        __syncthreads();
    }
#pragma unroll
    for (int i = 0; i < 2; i++)
#pragma unroll
        for (int j = 0; j < 2; j++) {
            int r0 = br + wr + i * 16 + rhalf;
            int c0 = bc + wc + j * 16 + m16;
#pragma unroll
            for (int v = 0; v < 8; v++)
                C[(size_t)(r0 + v) * ldc + c0] = acc[i][j][v];
        }
}

// ---------------------------------------------------------------------------
// 3) Apply 4x4 two-qubit gate in place on theta (stored as M: 768x768 row-major,
//    M[a*2+k, l*CHI+c]).  new[i,j] = sum_{k,l} G[(i2+j),(k2+l)] * old[k,l]
// ---------------------------------------------------------------------------
__global__ void apply_gate2_kernel(float* __restrict__ Mg, const float* __restrict__ G, int total) {
    int idx = blockIdx.x * blockDim.x + threadIdx.x;
    if (idx >= total) return;                  // total = CHI*CHI
    int a = idx / CHI, c = idx - (idx / CHI) * CHI;
    float x[4], o[4];
#pragma unroll
    for (int k = 0; k < 2; k++)
#pragma unroll
        for (int l = 0; l < 2; l++)
            x[k * 2 + l] = Mg[(size_t)(a * 2 + k) * TWO_CHI + l * CHI + c];
#pragma unroll
    for (int i = 0; i < 4; i++) {
        float s = 0.f;
#pragma unroll
        for (int k = 0; k < 4; k++) s += G[i * 4 + k] * x[k];
        o[i] = s;
    }
#pragma unroll
    for (int i = 0; i < 2; i++)
#pragma unroll
        for (int j = 0; j < 2; j++)
            Mg[(size_t)(a * 2 + i) * TWO_CHI + j * CHI + c] = o[i * 2 + j];
}

// ---------------------------------------------------------------------------
// 4) Init V = I and reset the software grid-barrier state
// ---------------------------------------------------------------------------
__global__ void init_v_bar_kernel(float* __restrict__ Vm, unsigned* __restrict__ bar, int total) {
    int idx = blockIdx.x * blockDim.x + threadIdx.x;
    if (idx == 0) { bar[0] = 0u; bar[1] = 0u; }
    if (idx >= total) return;                  // total = 768*768
    int i = idx / TWO_CHI, j = idx - i * TWO_CHI;
    Vm[idx] = (i == j) ? 1.0f : 0.0f;
}

// ---------------------------------------------------------------------------
// 5) Persistent parallel cyclic-Jacobi eigensolver for symmetric 768x768.
//    Software grid barrier (generation counter + s_sleep spin).
// ---------------------------------------------------------------------------
__device__ __forceinline__ void grid_barrier(unsigned* cnt, volatile unsigned* gen, int nblk) {
    __syncthreads();
    if (threadIdx.x == 0) {
        __threadfence();
        unsigned g = *gen;
        if (atomicAdd(cnt, 1u) == (unsigned)(nblk - 1)) {
            atomicExch(cnt, 0u);
            __threadfence();
            atomicAdd((unsigned*)gen, 1u);
        } else {
            while (*gen == g) { __builtin_amdgcn_s_sleep(1); }
        }
        __threadfence();
    }
    __syncthreads();
}

__global__ __launch_bounds__(256)
void jacobi_persistent_kernel(float* __restrict__ A, float* __restrict__ Vm,
                              float* cs, int* pq,
                              unsigned* barCnt, unsigned* barGen,
                              int nblk, int sweeps) {
    const int n = TWO_CHI;          // 768
    const int m = n - 1;            // 767 rounds (tournament schedule)
    const int npairs = n / 2;       // 384 disjoint pairs per round
    const int T = nblk * 256;
    const int gtid = blockIdx.x * 256 + threadIdx.x;

    for (int sw = 0; sw < sweeps; ++sw) {
        for (int r = 0; r < m; ++r) {
            // --- phase A: rotation angles from diagonal 2x2 blocks ---------
            if (gtid < npairs) {
                int k = gtid, p, q;
                if (k == 0) { p = m; q = r % m; }
                else        { p = (r + k) % m; q = (r + m - k) % m; }
                if (p > q) { int t = p; p = q; q = t; }
                float app = A[(size_t)p * n + p];
                float aqq = A[(size_t)q * n + q];
                float apq = A[(size_t)p * n + q];
                float c = 1.0f, s = 0.0f;
                if (fabsf(apq) > 1e-28f) {
                    float tau = (aqq - app) / (2.0f * apq);
                    float t = ((tau >= 0.0f) ? 1.0f : -1.0f) /
                              (fabsf(tau) + sqrtf(1.0f + tau * tau));
                    c = 1.0f / sqrtf(1.0f + t * t);
                    s = t * c;
                }
                cs[2 * k] = c; cs[2 * k + 1] = s;
                pq[2 * k] = p; pq[2 * k + 1] = q;
            }
            grid_barrier(barCnt, barGen, nblk);
            // --- phase B: row updates (A <- J^T A) -------------------------
            for (int w = gtid; w < npairs * n; w += T) {
                int k = w / n, j = w - (w / n) * n;
                int p = pq[2 * k], q = pq[2 * k + 1];
                float c = cs[2 * k], s = cs[2 * k + 1];
                float ap = A[(size_t)p * n + j], aq = A[(size_t)q * n + j];
                A[(size_t)p * n + j] = c * ap - s * aq;
                A[(size_t)q * n + j] = s * ap + c * aq;
            }
            grid_barrier(barCnt, barGen, nblk);
            // --- phase C: column updates (A <- A J, V <- V J) --------------
            for (int w = gtid; w < npairs * n; w += T) {
                int k = w / n, i = w - (w / n) * n;
                int p = pq[2 * k], q = pq[2 * k + 1];
                float c = cs[2 * k], s = cs[2 * k + 1];
                float ap = A[(size_t)i * n + p], aq = A[(size_t)i * n + q];
                A[(size_t)i * n + p] = c * ap - s * aq;
                A[(size_t)i * n + q] = s * ap + c * aq;
                float vp = Vm[(size_t)i * n + p], vq = Vm[(size_t)i * n + q];
                Vm[(size_t)i * n + p] = c * vp - s * vq;
                Vm[(size_t)i * n + q] = s * vp + c * vq;
            }
            grid_barrier(barCnt, barGen, nblk);
        }
    }
}

// ---------------------------------------------------------------------------
// 6) Bitonic sort of eigenvalues (descending) -> sorted indices + sigma=sqrt
// ---------------------------------------------------------------------------
__global__ __launch_bounds__(1024)
void sort_eigs_kernel(const float* __restrict__ A, int* __restrict__ sidx,
                      float* __restrict__ sigma) {
    __shared__ float sv[1024];
    __shared__ int   si[1024];
    int i = threadIdx.x;
    sv[i] = (i < TWO_CHI) ? A[(size_t)i * TWO_CHI + i] : -3.402823e38f;
    si[i] = i;
    __syncthreads();
    for (unsigned k = 2; k <= 1024; k <<= 1) {
        for (unsigned j = k >> 1; j > 0; j >>= 1) {
            unsigned ixj = (unsigned)i ^ j;
            if (ixj > (unsigned)i) {
                bool desc = (((unsigned)i & k) == 0);
                float vi = sv[i], vj = sv[ixj];
                bool sw = desc ? (vi < vj) : (vi > vj);
                if (sw) {
                    sv[i] = vj; sv[ixj] = vi;
                    int t = si[i]; si[i] = si[ixj]; si[ixj] = t;
                }
            }
            __syncthreads();
        }
    }
    if (i < TWO_CHI) sidx[i] = si[i];
    if (i < CHI)     sigma[i] = sqrtf(fmaxf(sv[i], 0.0f));
}

// ---------------------------------------------------------------------------
// 7) Gather top-CHI eigenvector columns (sorted)
// ---------------------------------------------------------------------------
__global__ void pack_vtop_kernel(const float* __restrict__ Vm, const int* __restrict__ sidx,
                                 float* __restrict__ Vtop, int total) {
    int idx = blockIdx.x * blockDim.x + threadIdx.x;
    if (idx >= total) return;                  // total = 768*384
    int i = idx / CHI, k = idx - (idx / CHI) * CHI;
    Vtop[(size_t)i * CHI + k] = Vm[(size_t)i * TWO_CHI + sidx[k]];
}

// ---------------------------------------------------------------------------
// 8) Writeback: siteB = U'/sigma (columns normalized), siteB1 = sigma * Vh
// ---------------------------------------------------------------------------
__global__ void writeback_kernel(const float* __restrict__ Up, const float* __restrict__ Vtop,
                                 const float* __restrict__ sigma,
                                 float* __restrict__ siteB, float* __restrict__ siteB1,
                                 int total) {
    int idx = blockIdx.x * blockDim.x + threadIdx.x;
    if (idx >= total) return;                  // total = 2*768*384
    if (idx < TWO_CHI * CHI) {
        int r = idx / CHI, k = idx - (idx / CHI) * CHI;
        float s = sigma[k];
        float inv = (s > 1e-12f) ? (1.0f / s) : 0.0f;
        siteB[(size_t)r * CHI + k] = Up[(size_t)r * CHI + k] * inv;
    } else {
        int t = idx - TWO_CHI * CHI;
        int k = t / TWO_CHI, mcol = t - (t / TWO_CHI) * TWO_CHI;
        siteB1[(size_t)k * TWO_CHI + mcol] = sigma[k] * Vtop[(size_t)mcol * CHI + k];
    }
}

// ---------------------------------------------------------------------------
extern "C" void kernel_launch(void* const* d_in, const int* in_sizes, int n_in,
                              void* d_out, int out_size, void* d_ws, size_t ws_size,
                              hipStream_t stream) {
    const float* mps0 = (const float*)d_in[0];   // (24, 384, 2, 384)
    const float* g1   = (const float*)d_in[1];   // (4, 24, 2, 2)
    const float* g2   = (const float*)d_in[2];   // (4, 23, 4, 4)
    float* out = (float*)d_out;                  // (24, 384, 2, 384)

    // workspace layout (floats)
    float* Mg    = (float*)d_ws;                 // 768*768 theta matrix
    float* Asym  = Mg   + TWO_CHI * TWO_CHI;     // 768*768 Gram
    float* Vm    = Asym + TWO_CHI * TWO_CHI;     // 768*768 eigenvectors
    float* Vtop  = Vm   + TWO_CHI * TWO_CHI;     // 768*384 sorted top vectors
    float* Up    = Vtop + TWO_CHI * CHI;         // 768*384 U' = M*Vtop
    float* sigma = Up   + TWO_CHI * CHI;         // 384
    float* cs    = sigma + CHI;                  // 768 rotation (c,s) pairs
    int*   pq    = (int*)(cs + TWO_CHI);         // 768 pair indices
    int*   sidx  = pq + TWO_CHI;                 // 768 sorted eig indices
    unsigned* bar = (unsigned*)(sidx + TWO_CHI); // [cnt, gen]

    for (int l = 0; l < NLAY; ++l) {
        const float* src = (l == 0) ? mps0 : out;
        {
            int total = NQ * CHI * CHI;
            mps_gate1_kernel<<<(total + 255) / 256, 256, 0, stream>>>(
                src, out, g1 + (size_t)l * NQ * 4, total);
        }
        for (int b = 0; b < NQ - 1; ++b) {
            float* siteB  = out + (size_t)b * SITE;
            float* siteB1 = out + (size_t)(b + 1) * SITE;
            // theta = A(768x384) * B(384x768)
            gemm_wmma_f32<<<dim3(TWO_CHI / TN, TWO_CHI / TM), 128, GEMM_LDS_BYTES, stream>>>(
                siteB, siteB1, Mg, TWO_CHI, TWO_CHI, CHI, CHI, TWO_CHI, TWO_CHI, 0);
            // apply 4x4 gate
            {
                int total = CHI * CHI;
                apply_gate2_kernel<<<(total + 255) / 256, 256, 0, stream>>>(
                    Mg, g2 + (size_t)(l * (NQ - 1) + b) * 16, total);
            }
            // Gram = M^T M (768x768)
            gemm_wmma_f32<<<dim3(TWO_CHI / TN, TWO_CHI / TM), 128, GEMM_LDS_BYTES, stream>>>(
                Mg, Mg, Asym, TWO_CHI, TWO_CHI, TWO_CHI, TWO_CHI, TWO_CHI, TWO_CHI, 1);
            // V = I, barrier reset
            {
                int total = TWO_CHI * TWO_CHI;
                init_v_bar_kernel<<<(total + 255) / 256, 256, 0, stream>>>(Vm, bar, total);
            }
            // Jacobi eigendecomposition of Gram
            jacobi_persistent_kernel<<<JNB, 256, 0, stream>>>(
                Asym, Vm, cs, pq, bar, bar + 1, JNB, NSWEEP);
            // sort eigenvalues, sigma = sqrt(top)
            sort_eigs_kernel<<<1, 1024, 0, stream>>>(Asym, sidx, sigma);
            // gather top-CHI eigenvectors
            {
                int total = TWO_CHI * CHI;
                pack_vtop_kernel<<<(total + 255) / 256, 256, 0, stream>>>(Vm, sidx, Vtop, total);
            }
            // U' = M * Vtop (768x384)
            gemm_wmma_f32<<<dim3(CHI / TN, TWO_CHI / TM), 128, GEMM_LDS_BYTES, stream>>>(
                Mg, Vtop, Up, TWO_CHI, CHI, TWO_CHI, TWO_CHI, CHI, CHI, 0);
            // write new site tensors
            {
                int total = 2 * TWO_CHI * CHI;
                writeback_kernel<<<(total + 255) / 256, 256, 0, stream>>>(
                    Up, Vtop, sigma, siteB, siteB1, total);
            }
        }
    }
}